// GAT_10754598109596
// MI455X (gfx1250) — compile-verified
//
#include <hip/hip_runtime.h>
#include <stdint.h>

#define B_  8
#define N_  2048
#define D_  128
#define NEGINF (-9e15f)
#define ALPHA_ 0.2f

typedef __attribute__((ext_vector_type(2))) float v2f;
typedef __attribute__((ext_vector_type(8))) float v8f;

// ---------------------------------------------------------------------------
// ASYNCcnt wait (async global->LDS copies)
// ---------------------------------------------------------------------------
__device__ __forceinline__ void wait_async0() {
#if __has_builtin(__builtin_amdgcn_s_wait_asynccnt)
  __builtin_amdgcn_s_wait_asynccnt(0);
#else
  asm volatile("s_wait_asynccnt 0x0" ::: "memory");
#endif
}

// Issue one per-lane 16B async copy from global memory at gbase plus byte offset goff into LDS at lds_ptr.
// Passing the LDS address through ptrtoint both supplies the VDST operand
// (low 32 bits of a generic LDS pointer == wave-relative LDS byte offset)
// and CAPTURES the shared array so the "memory" clobber covers it — without
// this, LLVM assumes the asm can't write the LDS array and folds the
// subsequent ds_loads to undef.
__device__ __forceinline__ void async_copy_b128(const void* lds_ptr, int goff,
                                                uint64_t gbase) {
  const uint32_t loff = (uint32_t)(uintptr_t)lds_ptr;
  asm volatile("global_load_async_to_lds_b128 %0, %1, %2"
               :: "v"(loff), "v"(goff), "s"(gbase)
               : "memory");
}

// ---------------------------------------------------------------------------
// Kernel 1: Wh = h @ W via V_WMMA_F32_16X16X4_F32.
// Block = 128 threads (4 waves). Wave w computes rows [blk*64 + w*16, +16).
// W (128x128 f32 = 64KB) staged in LDS via async global->LDS copies.
// ---------------------------------------------------------------------------
__global__ __launch_bounds__(128) void k_wh(const float* __restrict__ h,
                                            const float* __restrict__ W,
                                            float* __restrict__ Wh) {
  __shared__ float Wlds[D_ * D_];  // 64 KB
  const int tid = threadIdx.x;

  {
    const uint64_t wbase = (uint64_t)(uintptr_t)W;
#pragma unroll
    for (int it = 0; it < (D_ * D_ * 4) / (128 * 16); ++it) {  // 32 iters
      const int off = (tid + it * 128) * 16;
      async_copy_b128((const char*)Wlds + off, off, wbase);
    }
    wait_async0();
  }
  __syncthreads();

  const int wave = tid >> 5;
  const int lane = tid & 31;
  const int half = lane >> 4;   // 0: lanes 0-15, 1: lanes 16-31
  const int l16  = lane & 15;
  const int row0 = blockIdx.x * 64 + wave * 16;

  v8f acc[8];
#pragma unroll
  for (int t = 0; t < 8; ++t) acc[t] = (v8f)(0.0f);

  for (int k = 0; k < D_; k += 4) {
    // A fragment: 16x4 tile of h. VGPR0: K=k (lanes 0-15) / K=k+2 (16-31).
    v2f a;
    const float* hp = &h[(row0 + l16) * D_ + k + half * 2];
    a.x = hp[0];
    a.y = hp[1];
#pragma unroll
    for (int t = 0; t < 8; ++t) {
      // B fragment: 4x16 tile of W, rows striped across lanes per VGPR.
      v2f bf;
      bf.x = Wlds[(k + half) * D_ + t * 16 + l16];
      bf.y = Wlds[(k + half + 2) * D_ + t * 16 + l16];
      acc[t] = __builtin_amdgcn_wmma_f32_16x16x4_f32(
          false, a, false, bf, (short)0, acc[t], false, false);
    }
  }
  // C/D layout: lane holds column n = t*16 + l16, rows M = half*8 + r.
#pragma unroll
  for (int t = 0; t < 8; ++t) {
#pragma unroll
    for (int r = 0; r < 8; ++r) {
      Wh[(row0 + half * 8 + r) * D_ + t * 16 + l16] = acc[t][r];
    }
  }
}

// ---------------------------------------------------------------------------
// Kernel 2: Wh1 = Wh @ a[:d], Wh2 = Wh @ a[d:]. One wave per row, wave32
// shuffle reduction.
// ---------------------------------------------------------------------------
__global__ __launch_bounds__(256) void k_wh12(const float* __restrict__ Wh,
                                              const float* __restrict__ a,
                                              float* __restrict__ Wh1,
                                              float* __restrict__ Wh2) {
  const int wave = threadIdx.x >> 5;
  const int lane = threadIdx.x & 31;
  const int row  = blockIdx.x * 8 + wave;
  float s1 = 0.f, s2 = 0.f;
#pragma unroll
  for (int c = 0; c < 4; ++c) {
    const int n = lane + c * 32;
    const float v = Wh[row * D_ + n];
    s1 += v * a[n];
    s2 += v * a[D_ + n];
  }
#pragma unroll
  for (int off = 16; off > 0; off >>= 1) {
    s1 += __shfl_xor(s1, off, 32);
    s2 += __shfl_xor(s2, off, 32);
  }
  if (lane == 0) {
    Wh1[row] = s1;
    Wh2[row] = s2;
  }
}

// ---------------------------------------------------------------------------
// Kernel 3: masked leaky-relu softmax row. One block (256 thr) per (b,i).
// Writes attention straight into its d_out slice.
// ---------------------------------------------------------------------------
__global__ __launch_bounds__(256) void k_softmax(const int* __restrict__ adj,
                                                 const float* __restrict__ Wh1,
                                                 const float* __restrict__ Wh2,
                                                 float* __restrict__ att) {
  const int row = blockIdx.x;       // b*N + i
  const int b   = row / N_;
  const int tid = threadIdx.x;
  const long long base = (long long)row * N_;
  const float w1 = Wh1[row];
  const float* __restrict__ w2 = &Wh2[b * N_];
  const int* __restrict__ arow = &adj[base];

  __shared__ float red[8];

  float e[8];
  float m = NEGINF;
#pragma unroll
  for (int it = 0; it < 8; ++it) {
    const int j = tid + it * 256;
    float x = w1 + w2[j];
    x = (x >= 0.f) ? x : ALPHA_ * x;
    x = (arow[j] > 0) ? x : NEGINF;
    e[it] = x;
    m = fmaxf(m, x);
  }
#pragma unroll
  for (int off = 16; off > 0; off >>= 1) m = fmaxf(m, __shfl_xor(m, off, 32));
  if ((tid & 31) == 0) red[tid >> 5] = m;
  __syncthreads();
  m = red[0];
#pragma unroll
  for (int w = 1; w < 8; ++w) m = fmaxf(m, red[w]);
  __syncthreads();

  float s = 0.f;
#pragma unroll
  for (int it = 0; it < 8; ++it) {
    const float ex = __expf(e[it] - m);
    e[it] = ex;
    s += ex;
  }
#pragma unroll
  for (int off = 16; off > 0; off >>= 1) s += __shfl_xor(s, off, 32);
  if ((tid & 31) == 0) red[tid >> 5] = s;
  __syncthreads();
  s = red[0];
#pragma unroll
  for (int w = 1; w < 8; ++w) s += red[w];
  const float inv = 1.0f / s;
#pragma unroll
  for (int it = 0; it < 8; ++it) {
    att[base + tid + it * 256] = e[it] * inv;
  }
}

// ---------------------------------------------------------------------------
// Kernel 4: out = h + attention @ Wh via WMMA f32, K=2048 streamed through
// double-buffered 32x128 LDS slabs of Wh filled with async global->LDS
// copies so the DMA overlaps the WMMA block. Block = 128 thr -> 64 rows.
// ---------------------------------------------------------------------------
#define KK_ 32
__global__ __launch_bounds__(128) void k_hprime(const float* __restrict__ att,
                                                const float* __restrict__ Wh,
                                                const float* __restrict__ h,
                                                float* __restrict__ out) {
  __shared__ float Bs[2][KK_ * D_];   // 2 x 16 KB
  const int tid  = threadIdx.x;
  const int wave = tid >> 5;
  const int lane = tid & 31;
  const int half = lane >> 4;
  const int l16  = lane & 15;

  const int rowBlk = blockIdx.x * 64;
  const int b      = rowBlk / N_;
  const int row0   = rowBlk + wave * 16;
  const float* __restrict__ WhB = &Wh[(long long)b * N_ * D_];
  const uint64_t whbase = (uint64_t)(uintptr_t)WhB;

  // Issue async copy of K-slab [kk, kk+KK_) into buffer `buf`.
  auto issue_slab = [&](int buf, int kk) {
    const int lbase = buf * (KK_ * D_ * 4);
    const int gbase = kk * (D_ * 4);
#pragma unroll
    for (int it = 0; it < (KK_ * D_ * 4) / (128 * 16); ++it) {  // 8 iters
      const int o = (tid + it * 128) * 16;
      async_copy_b128((const char*)&Bs[0][0] + lbase + o, gbase + o, whbase);
    }
  };

  v8f acc[8];
#pragma unroll
  for (int t = 0; t < 8; ++t) acc[t] = (v8f)(0.0f);

  issue_slab(0, 0);
  wait_async0();
  __syncthreads();

  for (int kk = 0; kk < N_; kk += KK_) {
    const int cur = (kk / KK_) & 1;
    if (kk + KK_ < N_) issue_slab(cur ^ 1, kk + KK_);  // overlap with compute

    const float* __restrict__ Bcur = Bs[cur];
#pragma unroll
    for (int k4 = 0; k4 < KK_; k4 += 4) {
      v2f a;
      const float* ap =
          &att[(long long)(row0 + l16) * N_ + kk + k4 + half * 2];
      a.x = ap[0];
      a.y = ap[1];
#pragma unroll
      for (int t = 0; t < 8; ++t) {
        v2f bf;
        bf.x = Bcur[(k4 + half) * D_ + t * 16 + l16];
        bf.y = Bcur[(k4 + half + 2) * D_ + t * 16 + l16];
        acc[t] = __builtin_amdgcn_wmma_f32_16x16x4_f32(
            false, a, false, bf, (short)0, acc[t], false, false);
      }
    }
    wait_async0();      // next slab landed
    __syncthreads();    // everyone done with current slab + sees next one
  }

  // residual + store
#pragma unroll
  for (int t = 0; t < 8; ++t) {
#pragma unroll
    for (int r = 0; r < 8; ++r) {
      const long long o =
          (long long)(row0 + half * 8 + r) * D_ + t * 16 + l16;
      out[o] = h[o] + acc[t][r];
    }
  }
}

// ---------------------------------------------------------------------------
extern "C" void kernel_launch(void* const* d_in, const int* in_sizes, int n_in,
                              void* d_out, int out_size, void* d_ws, size_t ws_size,
                              hipStream_t stream) {
  const float* h   = (const float*)d_in[0];
  const int*   adj = (const int*)d_in[1];
  const float* W   = (const float*)d_in[2];
  const float* a   = (const float*)d_in[3];

  float* out = (float*)d_out;                               // [B,N,D]
  float* att = (float*)d_out + (size_t)B_ * N_ * D_;        // [B,N,N]

  float* Wh  = (float*)d_ws;                                // B*N*D floats
  float* Wh1 = Wh + (size_t)B_ * N_ * D_;                   // B*N floats
  float* Wh2 = Wh1 + (size_t)B_ * N_;                       // B*N floats

  k_wh     <<<(B_ * N_) / 64, 128, 0, stream>>>(h, W, Wh);
  k_wh12   <<<(B_ * N_) / 8, 256, 0, stream>>>(Wh, a, Wh1, Wh2);
  k_softmax<<< B_ * N_,       256, 0, stream>>>(adj, Wh1, Wh2, att);
  k_hprime <<<(B_ * N_) / 64, 128, 0, stream>>>(att, Wh, h, out);
}